// ConstrastiveLoss_5454608466398
// MI455X (gfx1250) — compile-verified
//
#include <hip/hip_runtime.h>

typedef __attribute__((ext_vector_type(2))) float v2f;
typedef __attribute__((ext_vector_type(8))) float v8f;

#define BATCH      4096
#define DIM        256
#define KWAVES     4          // waves per block, K dimension split
#define KSPAN      (DIM / KWAVES)   // 64 columns per wave
#define ROWS_PB    16         // rows per block (one WMMA M-tile)

// One block = 16 rows of x1/x2. Each of the 4 waves covers a 64-wide K slice.
// Per K-step of 4, lane L loads float2 x1/x2 at [row = L&15, col = kbase + 2*(L>>4)],
// which is simultaneously the correct A-matrix (16x4) and B-matrix (4x16) VGPR
// layout for V_WMMA_F32_16X16X4_F32. diag(A_x1 * B_x2) accumulates the row dots.
__global__ __launch_bounds__(KWAVES * 32)
void cosine_positives_wmma(const float* __restrict__ x1,
                           const float* __restrict__ x2,
                           float* __restrict__ out)
{
    __shared__ float part[KWAVES][ROWS_PB][3];   // per-wave partial s11,s12,s22

    const int tid  = threadIdx.x;       // 0..127
    const int wave = tid >> 5;          // 0..3
    const int lane = tid & 31;          // 0..31
    const int r0   = blockIdx.x * ROWS_PB;

    const int row  = lane & 15;               // M index this lane feeds
    const int koff = (lane >> 4) << 1;        // 0 or 2 (K sub-offset per half-wave)

    const size_t base = (size_t)(r0 + row) * DIM + (size_t)(wave * KSPAN + koff);
    const float* p1 = x1 + base;
    const float* p2 = x2 + base;

    v8f c11 = {};   // x1 . x1  (norms of x1 rows on the diagonal)
    v8f c12 = {};   // x1 . x2  (the cosine numerators on the diagonal)
    v8f c22 = {};   // x2 . x2

#pragma unroll
    for (int k = 0; k < KSPAN; k += 4) {
        v2f a = *(const v2f*)(p1 + k);   // serves as A (16x4) and B (4x16) of x1 tile
        v2f b = *(const v2f*)(p2 + k);   // serves as A and B of x2 tile
        // 8 args: (neg_a, A, neg_b, B, c_mod, C, reuse_a, reuse_b)
        c11 = __builtin_amdgcn_wmma_f32_16x16x4_f32(false, a, false, a,
                                                    (short)0, c11, false, false);
        c12 = __builtin_amdgcn_wmma_f32_16x16x4_f32(false, a, false, b,
                                                    (short)0, c12, false, false);
        c22 = __builtin_amdgcn_wmma_f32_16x16x4_f32(false, b, false, b,
                                                    (short)0, c22, false, false);
    }

    // Diagonal extraction from the 16x16 f32 C layout:
    //   VGPR j, lanes 0-15  -> C[M=j  ][N=lane]
    //   VGPR j, lanes 16-31 -> C[M=j+8][N=lane-16]
    // diag(m) for m=0..7  lives at lane m,    VGPR m
    // diag(m) for m=8..15 lives at lane m+16, VGPR m-8
    int m = -1, j = 0;
    if (lane < 8)        { m = lane;      j = lane;      }
    else if (lane >= 24) { m = lane - 16; j = lane - 24; }

    float d11 = 0.0f, d12 = 0.0f, d22 = 0.0f;
#pragma unroll
    for (int q = 0; q < 8; ++q) {
        if (q == j) { d11 = c11[q]; d12 = c12[q]; d22 = c22[q]; }
    }

    if (m >= 0) {
        part[wave][m][0] = d11;
        part[wave][m][1] = d12;
        part[wave][m][2] = d22;
    }
    __syncthreads();

    // Deterministic fixed-order reduction across the 4 K-slices, then finalize.
    if (tid < ROWS_PB) {
        float s11 = 0.0f, s12 = 0.0f, s22 = 0.0f;
#pragma unroll
        for (int w = 0; w < KWAVES; ++w) {
            s11 += part[w][tid][0];
            s12 += part[w][tid][1];
            s22 += part[w][tid][2];
        }
        const float eps = 1e-12f;
        float n1 = fmaxf(sqrtf(s11), eps);
        float n2 = fmaxf(sqrtf(s22), eps);
        float v  = s12 / (n1 * n2);
        int gr = r0 + tid;
        out[gr]         = v;   // sim_ij half
        out[gr + BATCH] = v;   // sim_ji half (identical by symmetry)
    }
}

extern "C" void kernel_launch(void* const* d_in, const int* in_sizes, int n_in,
                              void* d_out, int out_size, void* d_ws, size_t ws_size,
                              hipStream_t stream) {
    (void)in_sizes; (void)n_in; (void)out_size; (void)d_ws; (void)ws_size;
    const float* x1 = (const float*)d_in[0];
    const float* x2 = (const float*)d_in[1];
    float* out = (float*)d_out;

    dim3 grid(BATCH / ROWS_PB);      // 256 blocks
    dim3 block(KWAVES * 32);         // 128 threads = 4 wave32
    hipLaunchKernelGGL(cosine_positives_wmma, grid, block, 0, stream, x1, x2, out);
}